// MaxStateSuper_34892314313144
// MI455X (gfx1250) — compile-verified
//
#include <hip/hip_runtime.h>
#include <hip/hip_bf16.h>

typedef float v2f __attribute__((ext_vector_type(2)));
typedef float v8f __attribute__((ext_vector_type(8)));
typedef int   v4i __attribute__((ext_vector_type(4)));

#define DIM     2048
#define OUT_ELEMS 16777216ull   // 4*2048*2048
#define FLT_BIG 3.402823466e38f

// ---------------------------------------------------------------------------
// CDNA5 async global->LDS copy (ASYNCcnt-tracked), with sync fallback.
// Builtin signature (probe-confirmed via diagnostic):
//   void (v4i addrspace(1)* gsrc, v4i addrspace(3)* ldst, imm i32, imm i32)
// ---------------------------------------------------------------------------
#if __has_builtin(__builtin_amdgcn_global_load_async_to_lds_b128)
#define HAVE_ASYNC 1
#else
#define HAVE_ASYNC 0
#endif

typedef __attribute__((address_space(1))) v4i* gptr4;
typedef __attribute__((address_space(3))) v4i* lptr4;

__device__ __forceinline__ void cp16(const float* g, float* l) {
#if HAVE_ASYNC
    __builtin_amdgcn_global_load_async_to_lds_b128(
        (gptr4)(v4i*)g, (lptr4)(v4i*)l, 0, 0);
#else
    *(float4*)l = *(const float4*)g;
#endif
}

__device__ __forceinline__ void wait_cp() {
#if HAVE_ASYNC
#if __has_builtin(__builtin_amdgcn_s_wait_asynccnt)
    __builtin_amdgcn_s_wait_asynccnt(0);
#else
    asm volatile("s_wait_asynccnt 0x0" ::: "memory");
#endif
#endif
}

// ---------------------------------------------------------------------------
// Kernel A: c[m][n] = sum_k X[m][k] * W[4096+n][k]   (chunk 2 projection)
// 128x128 workgroup tile, 8 waves (4M x 2N), wave tile 32x64, fp32 WMMA
// 16x16x4. Double-buffered async LDS staging, row stride 36 floats
// (bank-conflict-free for the 16-row x k-pair fragment pattern, 16B aligned).
// Dynamic LDS: 2*(128+128)*36*4 = 73728 B.
// ---------------------------------------------------------------------------
#define ABUF (128 * 36)

__global__ __launch_bounds__(256) void gemm_c_kernel(
    const float* __restrict__ X, const float* __restrict__ Wm,
    float* __restrict__ ws_c) {
    extern __shared__ float smem[];
    float* lx = smem;              // 2 buffers of 128*36
    float* lw = smem + 2 * ABUF;   // 2 buffers of 128*36

    const int t    = threadIdx.x;
    const int lane = t & 31;
    const int w    = t >> 5;
    const int wM   = w & 3;
    const int wN   = w >> 2;
    const int lm   = lane & 15;
    const int kp   = (lane >> 4) << 1;
    const int mh   = (lane >> 4) << 3;
    const int m0   = blockIdx.x * 128;
    const int n0   = blockIdx.y * 128;

    v8f acc[2][4] = {};

    // prologue: stage k-chunk 0 into buffer 0
#pragma unroll
    for (int ph = 0; ph < 4; ++ph) {
        int idx = t + ph * 256, row = idx >> 3, c4 = (idx & 7) << 2;
        cp16(&X[(size_t)(m0 + row) * DIM + c4], &lx[row * 36 + c4]);
        cp16(&Wm[(size_t)(4096 + n0 + row) * DIM + c4], &lw[row * 36 + c4]);
    }

    for (int kc = 0; kc < DIM; kc += 32) {
        const int cur = (kc >> 5) & 1;
        wait_cp();
        __syncthreads();
        if (kc + 32 < DIM) {
            const int nb = cur ^ 1;
#pragma unroll
            for (int ph = 0; ph < 4; ++ph) {
                int idx = t + ph * 256, row = idx >> 3, c4 = (idx & 7) << 2;
                cp16(&X[(size_t)(m0 + row) * DIM + kc + 32 + c4],
                     &lx[nb * ABUF + row * 36 + c4]);
                cp16(&Wm[(size_t)(4096 + n0 + row) * DIM + kc + 32 + c4],
                     &lw[nb * ABUF + row * 36 + c4]);
            }
        }
        const float* clx = lx + cur * ABUF;
        const float* clw = lw + cur * ABUF;
#pragma unroll
        for (int kk = 0; kk < 32; kk += 4) {
            v2f af[2], bf[4];
#pragma unroll
            for (int i = 0; i < 2; ++i)
                af[i] = *(const v2f*)&clx[(wM * 32 + i * 16 + lm) * 36 + kk + kp];
#pragma unroll
            for (int j = 0; j < 4; ++j)
                bf[j] = *(const v2f*)&clw[(wN * 64 + j * 16 + lm) * 36 + kk + kp];
#pragma unroll
            for (int i = 0; i < 2; ++i)
#pragma unroll
                for (int j = 0; j < 4; ++j)
                    acc[i][j] = __builtin_amdgcn_wmma_f32_16x16x4_f32(
                        false, af[i], false, bf[j], (short)0, acc[i][j],
                        false, false);
        }
        __syncthreads();
    }

#pragma unroll
    for (int i = 0; i < 2; ++i)
#pragma unroll
        for (int j = 0; j < 4; ++j)
#pragma unroll
            for (int v = 0; v < 8; ++v) {
                int m = m0 + wM * 32 + i * 16 + mh + v;
                int n = n0 + wN * 64 + j * 16 + lm;
                ws_c[(size_t)m * DIM + n] = acc[i][j][v];
            }
}

// ---------------------------------------------------------------------------
// Kernel B: segmented running max along s (8 segments of 256 rows per batch);
// re-lays c and e into output permutation col' = hd*16 + head (coalesced
// consumption in the fused epilogue); records per-segment maxima.
// ---------------------------------------------------------------------------
__global__ __launch_bounds__(256) void scan_seg_kernel(
    const float* __restrict__ ws_c, float* __restrict__ ws_cp,
    float* __restrict__ ws_ep, float* __restrict__ segmax) {
    const int bid = blockIdx.x;
    const int b   = bid >> 6;
    const int seg = (bid >> 3) & 7;
    const int hdg = bid & 7;
    const int t   = threadIdx.x;
    const int hdl  = t >> 4;
    const int head = t & 15;
    const int col_nat  = head * 128 + hdg * 16 + hdl;  // head*128 + hd
    const int col_perm = hdg * 256 + t;                // hd*16 + head
    const int mbase = b * 2048 + seg * 256;

    float rm = -FLT_BIG;
    for (int r = 0; r < 256; ++r) {
        size_t mi = (size_t)(mbase + r) * DIM;
        float v = ws_c[mi + col_nat];
        rm = fmaxf(rm, v);
        ws_cp[mi + col_perm] = v;
        ws_ep[mi + col_perm] = rm;
    }
    segmax[(b * 8 + seg) * DIM + col_perm] = rm;
}

// ---------------------------------------------------------------------------
// Kernel C: apply carry from earlier segments to e; emit state output
// (state[b,hd,0,head] flat = b*2048 + hd*16 + head = b*2048 + col_perm).
// ---------------------------------------------------------------------------
__global__ __launch_bounds__(256) void scan_fix_kernel(
    float* __restrict__ ws_ep, const float* __restrict__ segmax,
    float* __restrict__ out) {
    const int bid = blockIdx.x;
    const int b   = bid >> 6;
    const int seg = (bid >> 3) & 7;
    const int hdg = bid & 7;
    const int col = hdg * 256 + threadIdx.x;

    float carry = -FLT_BIG;
    for (int sg = 0; sg < seg; ++sg)
        carry = fmaxf(carry, segmax[(b * 8 + sg) * DIM + col]);

    if (seg > 0) {
        for (int r = 0; r < 256; ++r) {
            size_t o = (size_t)(b * 2048 + seg * 256 + r) * DIM + col;
            ws_ep[o] = fmaxf(ws_ep[o], carry);
        }
    }
    if (seg == 7) {
        float st = fmaxf(carry, segmax[(b * 8 + 7) * DIM + col]);
        out[OUT_ELEMS + (size_t)b * 2048 + col] = st;
    }
}

// ---------------------------------------------------------------------------
// Kernel D: fused GEMM for chunks {a=0, b=1, d=3} sharing A fragments
// (12 WMMAs per 8 fragment loads per k-step) + elementwise combine with c,e
// + permuted (contiguous) store. 8 waves 2(M)x4(J), wave tile 32x32 x3.
// Double-buffered async staging. Dynamic LDS: 2*(64+384)*36*4 = 129024 B.
// ---------------------------------------------------------------------------
#define DBUFX (64 * 36)
#define DBUFW (384 * 36)

__global__ __launch_bounds__(256) void gemm_fused_kernel(
    const float* __restrict__ X, const float* __restrict__ Wm,
    const float* __restrict__ ws_cp, const float* __restrict__ ws_ep,
    const float* __restrict__ alph, float* __restrict__ out) {
    extern __shared__ float smem[];
    float* lx = smem;                // 2 buffers of 64*36
    float* lw = smem + 2 * DBUFX;    // 2 buffers of 384*36

    const int t    = threadIdx.x;
    const int lane = t & 31;
    const int w    = t >> 5;
    const int wM   = w & 1;
    const int wJ   = w >> 1;
    const int lm   = lane & 15;
    const int kp   = (lane >> 4) << 1;
    const int mh   = (lane >> 4) << 3;
    const int m0   = blockIdx.x * 64;
    const int jt   = blockIdx.y;       // 0..15
    const int hd0  = jt * 8;

    v8f acc[3][2][2] = {};

    // prologue: stage k-chunk 0 into buffer 0
#pragma unroll
    for (int ph = 0; ph < 2; ++ph) {
        int idx = t + ph * 256, row = idx >> 3, c4 = (idx & 7) << 2;
        cp16(&X[(size_t)(m0 + row) * DIM + c4], &lx[row * 36 + c4]);
    }
#pragma unroll
    for (int ph = 0; ph < 12; ++ph) {
        int idx = t + ph * 256, r = idx >> 3, c4 = (idx & 7) << 2;
        int q = r >> 7, j = r & 127;
        int qoff = (q == 0) ? 0 : (q == 1) ? 2048 : 6144;
        int grow = qoff + ((j & 15) << 7) + hd0 + (j >> 4);
        cp16(&Wm[(size_t)grow * DIM + c4], &lw[r * 36 + c4]);
    }

    for (int kc = 0; kc < DIM; kc += 32) {
        const int cur = (kc >> 5) & 1;
        wait_cp();
        __syncthreads();
        if (kc + 32 < DIM) {
            const int nb = cur ^ 1;
#pragma unroll
            for (int ph = 0; ph < 2; ++ph) {
                int idx = t + ph * 256, row = idx >> 3, c4 = (idx & 7) << 2;
                cp16(&X[(size_t)(m0 + row) * DIM + kc + 32 + c4],
                     &lx[nb * DBUFX + row * 36 + c4]);
            }
#pragma unroll
            for (int ph = 0; ph < 12; ++ph) {
                int idx = t + ph * 256, r = idx >> 3, c4 = (idx & 7) << 2;
                int q = r >> 7, j = r & 127;
                int qoff = (q == 0) ? 0 : (q == 1) ? 2048 : 6144;
                int grow = qoff + ((j & 15) << 7) + hd0 + (j >> 4);
                cp16(&Wm[(size_t)grow * DIM + kc + 32 + c4],
                     &lw[nb * DBUFW + r * 36 + c4]);
            }
        }
        const float* clx = lx + cur * DBUFX;
        const float* clw = lw + cur * DBUFW;
#pragma unroll
        for (int kk = 0; kk < 32; kk += 4) {
            v2f af[2], bf[3][2];
#pragma unroll
            for (int i = 0; i < 2; ++i)
                af[i] = *(const v2f*)&clx[(wM * 32 + i * 16 + lm) * 36 + kk + kp];
#pragma unroll
            for (int q = 0; q < 3; ++q)
#pragma unroll
                for (int jj = 0; jj < 2; ++jj)
                    bf[q][jj] = *(const v2f*)
                        &clw[((q << 7) + wJ * 32 + jj * 16 + lm) * 36 + kk + kp];
#pragma unroll
            for (int q = 0; q < 3; ++q)
#pragma unroll
                for (int i = 0; i < 2; ++i)
#pragma unroll
                    for (int jj = 0; jj < 2; ++jj)
                        acc[q][i][jj] = __builtin_amdgcn_wmma_f32_16x16x4_f32(
                            false, af[i], false, bf[q][jj], (short)0,
                            acc[q][i][jj], false, false);
        }
        __syncthreads();
    }

    const float a1 = alph[0], a2 = alph[1], a3 = alph[2];
#pragma unroll
    for (int i = 0; i < 2; ++i)
#pragma unroll
        for (int jj = 0; jj < 2; ++jj)
#pragma unroll
            for (int v = 0; v < 8; ++v) {
                int m = m0 + wM * 32 + i * 16 + mh + v;
                int j = wJ * 32 + jj * 16 + lm;
                int col = jt * 128 + j;           // = (hd0 + j/16)*16 + j%16
                size_t o = (size_t)m * DIM + col;
                float cc = ws_cp[o];
                float ee = ws_ep[o];
                float av = acc[0][i][jj][v];
                float bv = acc[1][i][jj][v];
                float dv = acc[2][i][jj][v];
                out[o] = av * bv + a1 * bv + a2 * dv +
                         av * (a3 * ee + dv) + bv * (cc + ee) + cc * ee;
            }
}

extern "C" void kernel_launch(void* const* d_in, const int* in_sizes, int n_in,
                              void* d_out, int out_size, void* d_ws, size_t ws_size,
                              hipStream_t stream) {
    const float* X    = (const float*)d_in[0];   // (4, 2048, 2048) fp32
    const float* Wm   = (const float*)d_in[1];   // (8192, 2048) fp32
    const float* alph = (const float*)d_in[2];   // (4,) fp32
    float* out = (float*)d_out;                  // 16777216 out + 8192 state

    float* ws_c   = (float*)d_ws;                // natural-layout c : 64 MiB
    float* ws_cp  = ws_c  + OUT_ELEMS;           // permuted c       : 64 MiB
    float* ws_ep  = ws_cp + OUT_ELEMS;           // permuted e       : 64 MiB
    float* segmax = ws_ep + OUT_ELEMS;           // 4*8*2048 floats

    const size_t ldsA = 4ull * ABUF * sizeof(float);             // 73728 B
    const size_t ldsD = 2ull * (DBUFX + DBUFW) * sizeof(float);  // 129024 B

    gemm_c_kernel<<<dim3(64, 16), 256, ldsA, stream>>>(X, Wm, ws_c);
    scan_seg_kernel<<<256, 256, 0, stream>>>(ws_c, ws_cp, ws_ep, segmax);
    scan_fix_kernel<<<256, 256, 0, stream>>>(ws_ep, segmax, out);
    gemm_fused_kernel<<<dim3(128, 16), 256, ldsD, stream>>>(X, Wm, ws_cp, ws_ep,
                                                            alph, out);
}